// ImprovedGATNet_18322330485178
// MI455X (gfx1250) — compile-verified
//
#include <hip/hip_runtime.h>
#include <hip/hip_bf16.h>
#include <math.h>

typedef float v2f __attribute__((ext_vector_type(2)));
typedef float v8f __attribute__((ext_vector_type(8)));

#define HIDC 128
#define BN_EPS 1e-5f

__device__ inline float leaky02(float x) { return x > 0.0f ? x : 0.2f * x; }

// ---------------------------------------------------------------------------
// FP32 WMMA GEMM, row-strip per wave: one wave owns a 16-row strip of C and
// sweeps TN 16-col tiles with TN accumulators. A fragment loaded once per
// K-step (v_wmma_f32_16x16x4_f32, K stepped by 4); B stays L2/WGP$-resident.
// Epilogue: v = act(x + bias); v = v*bn_scale + bn_shift  (ReLU before BN,
// matching reference order). M multiple of 16 (50000 = 3125*16).
// ---------------------------------------------------------------------------
template <int TN>
__global__ __launch_bounds__(256) void gemm_f32_wmma(
    const float* __restrict__ A, const float* __restrict__ B,
    float* __restrict__ C, int M, int K,
    const float* __restrict__ bias,
    const float* __restrict__ bn_g, const float* __restrict__ bn_b,
    int act) {
  const int Nc  = TN * 16;
  int wid  = (blockIdx.x * blockDim.x + threadIdx.x) >> 5;  // row-strip id
  int lane = threadIdx.x & 31;
  if (wid >= (M >> 4)) return;                              // wave-uniform
  int half = lane >> 4;            // 0: lanes 0-15, 1: lanes 16-31
  int l16  = lane & 15;
  int arow = wid * 16 + l16;

  const float* ap = A + (size_t)arow * K;
  v8f zero = {0.f, 0.f, 0.f, 0.f, 0.f, 0.f, 0.f, 0.f};
  v8f acc[TN];
  #pragma unroll
  for (int t = 0; t < TN; ++t) acc[t] = zero;

  for (int k0 = 0; k0 < K; k0 += 4) {
    int ka = k0 + half * 2;        // A: lanes0-15 K=k0,k0+1; lanes16-31 K=k0+2,k0+3
    v2f a; a.x = ap[ka]; a.y = ap[ka + 1];
    const float* bp0 = B + (size_t)ka * Nc + l16;
    const float* bp1 = bp0 + Nc;
    #pragma unroll
    for (int t = 0; t < TN; ++t) {
      v2f b; b.x = bp0[t * 16]; b.y = bp1[t * 16];
      acc[t] = __builtin_amdgcn_wmma_f32_16x16x4_f32(
          false, a, false, b, (short)0, acc[t], false, false);
    }
  }

  #pragma unroll
  for (int t = 0; t < TN; ++t) {
    int bcol = t * 16 + l16;
    float bi = bias ? bias[bcol] : 0.0f;
    float sc = 1.0f, sh = 0.0f;
    if (bn_g) { sc = bn_g[bcol] * rsqrtf(1.0f + BN_EPS); sh = bn_b[bcol]; }
    #pragma unroll
    for (int r = 0; r < 8; ++r) {  // C/D layout: vgpr r -> M=r (+8 upper half)
      int orow = wid * 16 + r + half * 8;
      float v = acc[t][r] + bi;
      if (act >= 1) v = fmaxf(v, 0.0f);
      v = v * sc + sh;
      C[(size_t)orow * Nc + bcol] = v;
    }
  }
}

// ---------------------------------------------------------------------------
// Per-node attention logits: als[n,h] = sum_c xh[n,h*C+c]*a_s[h,c]
// ---------------------------------------------------------------------------
__global__ __launch_bounds__(128) void gat_logits(
    const float* __restrict__ xh, const float* __restrict__ a_s,
    const float* __restrict__ a_d, float* __restrict__ als,
    float* __restrict__ ald, int H, int Cc) {
  int n = blockIdx.x;
  int t = threadIdx.x;
  __shared__ float ss[HIDC], sd[HIDC];
  float v = xh[(size_t)n * HIDC + t];
  ss[t] = v * a_s[t];
  sd[t] = v * a_d[t];
  __syncthreads();
  if (t < H) {
    float a = 0.f, b = 0.f;
    for (int c = 0; c < Cc; ++c) { a += ss[t * Cc + c]; b += sd[t * Cc + c]; }
    als[n * H + t] = a;
    ald[n * H + t] = b;
  }
}

// ---------------------------------------------------------------------------
// CSR build (dst -> src), done ONCE: topology is identical for all 3 layers.
// ---------------------------------------------------------------------------
__device__ inline void edge_sd(const int* __restrict__ ei, int e, int E,
                               int& s, int& d) {
  if (e < E) { s = ei[e]; d = ei[E + e]; }   // edge_index rows [2,E]
  else       { s = d = e - E; }               // self loops
}

__global__ void csr_zero(int* __restrict__ deg, int* __restrict__ cursor, int N) {
  int i = blockIdx.x * blockDim.x + threadIdx.x;
  if (i < N) { deg[i] = 0; cursor[i] = 0; }
}

__global__ void csr_count(const int* __restrict__ ei, int* __restrict__ deg,
                          int E, int N) {
  int e = blockIdx.x * blockDim.x + threadIdx.x;
  if (e >= E + N) return;
  int s, d; edge_sd(ei, e, E, s, d);
  atomicAdd(&deg[d], 1);
}

// inclusive block scan -> tmp, block totals -> bsum
__global__ __launch_bounds__(256) void scan_block(const int* __restrict__ deg,
                                                  int* __restrict__ tmp,
                                                  int* __restrict__ bsum, int N) {
  __shared__ int sm[256];
  int i = blockIdx.x * 256 + threadIdx.x;
  sm[threadIdx.x] = (i < N) ? deg[i] : 0;
  __syncthreads();
  for (int o = 1; o < 256; o <<= 1) {
    int add = (threadIdx.x >= o) ? sm[threadIdx.x - o] : 0;
    __syncthreads();
    sm[threadIdx.x] += add;
    __syncthreads();
  }
  if (i < N) tmp[i] = sm[threadIdx.x];
  if (threadIdx.x == 255) bsum[blockIdx.x] = sm[255];
}

// single-block inclusive scan of block totals (nb <= 256)
__global__ __launch_bounds__(256) void scan_top(int* __restrict__ bsum, int nb) {
  __shared__ int sm[256];
  int t = threadIdx.x;
  sm[t] = (t < nb) ? bsum[t] : 0;
  __syncthreads();
  for (int o = 1; o < 256; o <<= 1) {
    int add = (t >= o) ? sm[t - o] : 0;
    __syncthreads();
    sm[t] += add;
    __syncthreads();
  }
  if (t < nb) bsum[t] = sm[t];
}

// off[0]=0; off[i+1]=inclusive_sum(i)  -> exclusive offsets
__global__ void scan_add(const int* __restrict__ tmp, const int* __restrict__ bsum,
                         int* __restrict__ off, int N) {
  int i = blockIdx.x * 256 + threadIdx.x;
  if (i == 0) off[0] = 0;
  if (i < N) off[i + 1] = tmp[i] + (blockIdx.x > 0 ? bsum[blockIdx.x - 1] : 0);
}

__global__ void csr_fill(const int* __restrict__ ei, const int* __restrict__ off,
                         int* __restrict__ cursor, int* __restrict__ csr,
                         int E, int N) {
  int e = blockIdx.x * blockDim.x + threadIdx.x;
  if (e >= E + N) return;
  int s, d; edge_sd(ei, e, E, s, d);
  int pos = off[d] + atomicAdd(&cursor[d], 1);
  csr[pos] = s;
}

// ---------------------------------------------------------------------------
// Fused GAT aggregate per dst node (no atomics): one 128-thread block per node.
// Phase 1: waves 0..H-1 compute per-head segment max & softmax denominator
// via __shfl_xor wave reductions. Phase 2: every thread owns one output
// channel and accumulates sum_e w_e * xh[src_e, c] with coalesced gathers.
// Epilogue: elu(bn(acc + bias)) [* fi].
// ---------------------------------------------------------------------------
__global__ __launch_bounds__(128) void gat_fused_aggr(
    const int* __restrict__ off, const int* __restrict__ csr,
    const float* __restrict__ xh, const float* __restrict__ als,
    const float* __restrict__ ald, const float* __restrict__ bias,
    const float* __restrict__ g, const float* __restrict__ be,
    const float* __restrict__ fi, float* __restrict__ outp,
    int H, int Cc, int useFi) {
  int n = blockIdx.x;
  int t = threadIdx.x;
  int lane = t & 31, wv = t >> 5;
  int e0 = off[n], e1 = off[n + 1];
  __shared__ float s_max[4], s_den[4];

  if (wv < H) {                       // every node has >=1 edge (self-loop)
    int h = wv;
    float aldn = ald[n * H + h];
    float mx = -__builtin_inff();
    for (int e = e0 + lane; e < e1; e += 32)
      mx = fmaxf(mx, leaky02(als[csr[e] * H + h] + aldn));
    #pragma unroll
    for (int o = 16; o; o >>= 1) mx = fmaxf(mx, __shfl_xor(mx, o, 32));
    float sum = 0.0f;
    for (int e = e0 + lane; e < e1; e += 32)
      sum += expf(leaky02(als[csr[e] * H + h] + aldn) - mx);
    #pragma unroll
    for (int o = 16; o; o >>= 1) sum += __shfl_xor(sum, o, 32);
    if (lane == 0) { s_max[h] = mx; s_den[h] = sum; }
  }
  __syncthreads();

  int h = t / Cc;
  float aldn = ald[n * H + h];
  float mx = s_max[h];
  float inv_dn = 1.0f / s_den[h];
  float acc = 0.0f;
  for (int e = e0; e < e1; ++e) {
    int s = csr[e];
    float w = expf(leaky02(als[s * H + h] + aldn) - mx) * inv_dn;
    acc += xh[(size_t)s * HIDC + t] * w;
  }

  float v = acc + bias[t];
  v = v * (g[t] * rsqrtf(1.0f + BN_EPS)) + be[t];
  v = v > 0.0f ? v : (expf(v) - 1.0f);    // ELU
  if (useFi) v *= fi[t];
  outp[(size_t)n * HIDC + t] = v;
}

// final [N,32] @ [32,2] + bc3
__global__ void final_lin(const float* __restrict__ h, const float* __restrict__ W,
                          const float* __restrict__ b, float* __restrict__ out, int N) {
  int n = blockIdx.x * blockDim.x + threadIdx.x;
  if (n >= N) return;
  float o0 = b[0], o1 = b[1];
  const float* hp = h + (size_t)n * 32;
  #pragma unroll
  for (int k = 0; k < 32; ++k) {
    float v = hp[k];
    o0 += v * W[k * 2 + 0];
    o1 += v * W[k * 2 + 1];
  }
  out[n * 2 + 0] = o0;
  out[n * 2 + 1] = o1;
}

extern "C" void kernel_launch(void* const* d_in, const int* in_sizes, int n_in,
                              void* d_out, int out_size, void* d_ws, size_t ws_size,
                              hipStream_t stream) {
  const float* x  = (const float*)d_in[0];
  const int*   ei = (const int*)d_in[1];
  const float* Wp = (const float*)d_in[2];
  const float* bp = (const float*)d_in[3];
  const float* Wl[3]  = {(const float*)d_in[4],  (const float*)d_in[10], (const float*)d_in[16]};
  const float* asl[3] = {(const float*)d_in[5],  (const float*)d_in[11], (const float*)d_in[17]};
  const float* adl[3] = {(const float*)d_in[6],  (const float*)d_in[12], (const float*)d_in[18]};
  const float* bl[3]  = {(const float*)d_in[7],  (const float*)d_in[13], (const float*)d_in[19]};
  const float* gl[3]  = {(const float*)d_in[8],  (const float*)d_in[14], (const float*)d_in[20]};
  const float* bel[3] = {(const float*)d_in[9],  (const float*)d_in[15], (const float*)d_in[21]};
  const float* fi  = (const float*)d_in[22];
  const float* Wc1 = (const float*)d_in[23];
  const float* bc1 = (const float*)d_in[24];
  const float* gc  = (const float*)d_in[25];
  const float* bec = (const float*)d_in[26];
  const float* Wc2 = (const float*)d_in[27];
  const float* bc2 = (const float*)d_in[28];
  const float* Wc3 = (const float*)d_in[29];
  const float* bc3 = (const float*)d_in[30];
  float* out = (float*)d_out;

  const int N  = in_sizes[0] / 64;   // 50000
  const int E  = in_sizes[1] / 2;    // 800000
  const int E2 = E + N;

  // --- workspace layout ---
  float* ws   = (float*)d_ws;
  float* hbuf = ws;                        // N*128 node features (ping)
  float* xhb  = hbuf + (size_t)N * HIDC;   // N*128 xh / classifier c1
  float* als  = xhb  + (size_t)N * HIDC;   // N*4
  float* ald  = als  + (size_t)N * 4;      // N*4
  int* deg    = (int*)(ald + (size_t)N * 4);   // N
  int* cursor = deg + N;                   // N
  int* off    = cursor + N;                // N+1
  int* bsum   = off + (N + 1);             // <=256
  int* tmp    = bsum + 256;                // N
  int* csr    = tmp + N;                   // E2

  auto gemm = [&](const float* A, const float* B, float* C, int M, int Nc, int K,
                  const float* bias, const float* g2, const float* b2, int act) {
    int blocks = ((M / 16) + 7) / 8;       // 8 row-strip waves per 256-thr block
    if (Nc == 128)
      gemm_f32_wmma<8><<<blocks, 256, 0, stream>>>(A, B, C, M, K, bias, g2, b2, act);
    else if (Nc == 64)
      gemm_f32_wmma<4><<<blocks, 256, 0, stream>>>(A, B, C, M, K, bias, g2, b2, act);
    else
      gemm_f32_wmma<2><<<blocks, 256, 0, stream>>>(A, B, C, M, K, bias, g2, b2, act);
  };

  // --- build CSR once (same topology for all layers) ---
  int nbN = (N + 255) / 256;
  int nbE = (E2 + 255) / 256;
  csr_zero<<<nbN, 256, 0, stream>>>(deg, cursor, N);
  csr_count<<<nbE, 256, 0, stream>>>(ei, deg, E, N);
  scan_block<<<nbN, 256, 0, stream>>>(deg, tmp, bsum, N);
  scan_top<<<1, 256, 0, stream>>>(bsum, nbN);
  scan_add<<<nbN, 256, 0, stream>>>(tmp, bsum, off, N);
  csr_fill<<<nbE, 256, 0, stream>>>(ei, off, cursor, csr, E, N);

  // --- projection: h = x @ Wp + bp ---
  gemm(x, Wp, hbuf, N, HIDC, 64, bp, nullptr, nullptr, 0);

  // --- three GAT layers ---
  const int Hh[3] = {4, 4, 1};
  for (int l = 0; l < 3; ++l) {
    int H = Hh[l], Cc = HIDC / H;
    gemm(hbuf, Wl[l], xhb, N, HIDC, HIDC, nullptr, nullptr, nullptr, 0);
    gat_logits<<<N, HIDC, 0, stream>>>(xhb, asl[l], adl[l], als, ald, H, Cc);
    gat_fused_aggr<<<N, HIDC, 0, stream>>>(off, csr, xhb, als, ald,
                                           bl[l], gl[l], bel[l], fi, hbuf,
                                           H, Cc, (l == 2) ? 1 : 0);
  }

  // --- classifier head ---
  gemm(hbuf, Wc1, xhb, N, 64, HIDC, bc1, gc, bec, 1);   // bn(relu(.))
  gemm(xhb, Wc2, hbuf, N, 32, 64, bc2, nullptr, nullptr, 1);  // relu(.)
  final_lin<<<(N + 255) / 256, 256, 0, stream>>>(hbuf, Wc3, bc3, out, N);
}